// DGCNN_30339648979404
// MI455X (gfx1250) — compile-verified
//
#include <hip/hip_runtime.h>
#include <hip/hip_bf16.h>

// ---------------------------------------------------------------------------
// DGCNN forward for MI455X (gfx1250, wave32, WMMA f32 16x16x4).
//
//  per layer li in 0..3:
//    li>0: sqnorm -> per-batch WMMA GEMM negdist = 2*H*H^T - sq_i - sq_j
//          (register-blocked 16x64 per wave) -> wave32 top-20 per row
//    W = [Wd | Wi];  Wc = Wi - Wd  (zero-padded so k-loops are exact)
//    Dp = H * Wd^T ;  Ap = H * Wc^T + b     (WMMA GEMM over B*N rows)
//    F[:, off:off+Co] = max_k lrelu(g*inv*(Dp[idx_k] + Ap) + be)  (L2 gather)
//  final: out[b,o] = max_n lrelu(F[b,n,:512] . Wf[o,:] + bf[o])
//         (fused WMMA, 64x16 per wave per iteration, running max in regs)
// ---------------------------------------------------------------------------

typedef __attribute__((ext_vector_type(2))) float v2f;
typedef __attribute__((ext_vector_type(8))) float v8f;

#define BATCH   8
#define NPTS    4096
#define KNN     20
#define LDFEAT  512
#define INV_BN  0.9999950000374997f   // 1/sqrt(1+1e-5)
#define SLOPE   0.2f
#define NEG_INF (-3.4e38f)

__device__ __forceinline__ float lrelu(float x) { return x > 0.f ? x : SLOPE * x; }

#define WMMA_F32(a, b, c) \
  __builtin_amdgcn_wmma_f32_16x16x4_f32(false, (a), false, (b), (short)0, (c), false, false)

// ---------------------------------------------------------------------------
// Single 16x16 tile (used by proj kernels; small-FLOP GEMMs).
// ---------------------------------------------------------------------------
template <int C>
__device__ __forceinline__ v8f wmma_nt_16x16(const float* __restrict__ A, int lda,
                                             const float* __restrict__ Bt, int ldb,
                                             int m0, int n0) {
  const int lane = threadIdx.x & 31;
  const int koff = (lane >> 4) * 2;
  const int l16  = lane & 15;
  const float* arow = A  + (long)(m0 + l16) * lda + koff;
  const float* brow = Bt + (long)(n0 + l16) * ldb + koff;
  v8f acc = {0.f, 0.f, 0.f, 0.f, 0.f, 0.f, 0.f, 0.f};
#pragma unroll 4
  for (int k0 = 0; k0 < C; k0 += 4) {
    v2f a = *(const v2f*)(arow + k0);
    v2f b = *(const v2f*)(brow + k0);
    acc = WMMA_F32(a, b, acc);
  }
  return acc;
}

// ---------------------------------------------------------------------------
// Pad x [B*N,3] -> Xp [B*N,4] so layer-0 GEMM k-loops need no guards.
// ---------------------------------------------------------------------------
__global__ void padx_kernel(const float* __restrict__ x, float* __restrict__ Xp,
                            int total) {
  int i = blockIdx.x * blockDim.x + threadIdx.x;
  if (i >= total * 4) return;
  int c = i & 3, r = i >> 2;
  Xp[i] = (c < 3) ? x[r * 3 + c] : 0.f;
}

// ---------------------------------------------------------------------------
// Squared norms of feature rows.
// ---------------------------------------------------------------------------
__global__ void sqnorm_kernel(const float* __restrict__ H, int ldh,
                              float* __restrict__ sqn, int total, int C) {
  int i = blockIdx.x * blockDim.x + threadIdx.x;
  if (i >= total) return;
  const float* r = H + (long)i * ldh;
  float s = 0.f;
  for (int c = 0; c < C; ++c) s += r[c] * r[c];
  sqn[i] = s;
}

// ---------------------------------------------------------------------------
// negdist = 2*H*H^T - sq_i - sq_j.  Register-blocked: one wave computes a
// 16(m) x 64(n) strip with 4 accumulators; per k-step 1 A-load + 4 B-loads
// feed 4 WMMAs (1.25 loads/WMMA vs 2 unblocked).
// ---------------------------------------------------------------------------
template <int C>
__global__ __launch_bounds__(256)
void knn_dist_kernel(const float* __restrict__ H, int ldh,
                     const float* __restrict__ sqn,
                     float* __restrict__ dist, int N) {
  const int tilesM  = N >> 4;
  const int tilesN4 = N >> 6;                       // groups of 4 n-tiles
  const int wave = blockIdx.x * (blockDim.x >> 5) + (threadIdx.x >> 5);
  if (wave >= tilesM * tilesN4) return;             // wave-uniform exit
  const int tm = wave / tilesN4, tg = wave % tilesN4;
  const int m0 = tm * 16, n0 = tg * 64;

  const int lane = threadIdx.x & 31;
  const int koff = (lane >> 4) * 2;
  const int l16  = lane & 15;
  const float* arow = H + (long)(m0 + l16) * ldh + koff;
  const float* b0   = H + (long)(n0 + l16) * ldh + koff;
  const float* b1   = b0 + 16 * ldh;
  const float* b2   = b1 + 16 * ldh;
  const float* b3   = b2 + 16 * ldh;

  v8f c0 = {0.f,0.f,0.f,0.f,0.f,0.f,0.f,0.f};
  v8f c1 = c0, c2 = c0, c3 = c0;
#pragma unroll 2
  for (int k0 = 0; k0 < C; k0 += 4) {
    v2f a  = *(const v2f*)(arow + k0);
    v2f q0 = *(const v2f*)(b0 + k0);
    v2f q1 = *(const v2f*)(b1 + k0);
    v2f q2 = *(const v2f*)(b2 + k0);
    v2f q3 = *(const v2f*)(b3 + k0);
    c0 = WMMA_F32(a, q0, c0);
    c1 = WMMA_F32(a, q1, c1);
    c2 = WMMA_F32(a, q2, c2);
    c3 = WMMA_F32(a, q3, c3);
  }

  const int mbase = m0 + 8 * (lane >> 4);
  float sm[8];
#pragma unroll
  for (int v = 0; v < 8; ++v) sm[v] = sqn[mbase + v];

  const v8f* cc[4] = {&c0, &c1, &c2, &c3};
#pragma unroll
  for (int j = 0; j < 4; ++j) {
    const int n = n0 + j * 16 + l16;
    const float sn = sqn[n];
    const v8f& c = *cc[j];
#pragma unroll
    for (int v = 0; v < 8; ++v)
      dist[(long)(mbase + v) * N + n] = 2.f * c[v] - sm[v] - sn;
  }
}

// ---------------------------------------------------------------------------
// Per-row top-20 of negdist. One wave32 per row; per-lane register top-20,
// then 20 rounds of shfl_xor argmax merge (ties -> smaller index).
// ---------------------------------------------------------------------------
__global__ __launch_bounds__(256)
void topk_kernel(const float* __restrict__ dist, int N,
                 int* __restrict__ outIdx, int rowBase) {
  const int row = blockIdx.x * (blockDim.x >> 5) + (threadIdx.x >> 5);
  if (row >= N) return;
  const int lane = threadIdx.x & 31;
  const float* dr = dist + (long)row * N;

  float vals[KNN]; int idxs[KNN];
#pragma unroll
  for (int t = 0; t < KNN; ++t) { vals[t] = NEG_INF; idxs[t] = 0x7fffffff; }

  for (int j = lane; j < N; j += 32) {
    float v = dr[j];
    if (v > vals[KNN - 1]) {
      vals[KNN - 1] = v; idxs[KNN - 1] = j;
#pragma unroll
      for (int t = KNN - 1; t > 0; --t) {
        if (vals[t] > vals[t - 1] ||
            (vals[t] == vals[t - 1] && idxs[t] < idxs[t - 1])) {
          float tv = vals[t]; vals[t] = vals[t - 1]; vals[t - 1] = tv;
          int   ti = idxs[t]; idxs[t] = idxs[t - 1]; idxs[t - 1] = ti;
        }
      }
    }
  }

  int* orow = outIdx + (long)(rowBase + row) * KNN;
  for (int t = 0; t < KNN; ++t) {
    float bv = vals[0]; int bi = idxs[0]; int bl = lane;
#pragma unroll
    for (int off = 16; off > 0; off >>= 1) {
      float ov = __shfl_xor(bv, off, 32);
      int   oi = __shfl_xor(bi, off, 32);
      int   ol = __shfl_xor(bl, off, 32);
      if (ov > bv || (ov == bv && oi < bi)) { bv = ov; bi = oi; bl = ol; }
    }
    if (lane == bl) {
#pragma unroll
      for (int t2 = 0; t2 < KNN - 1; ++t2) { vals[t2] = vals[t2 + 1]; idxs[t2] = idxs[t2 + 1]; }
      vals[KNN - 1] = NEG_INF; idxs[KNN - 1] = 0x7fffffff;
    }
    if (lane == 0) orow[t] = bi;
  }
}

// ---------------------------------------------------------------------------
// Split + pad weights: Wdp[o,c] = W[o,c]; Wcp[o,c] = W[o,Ci+c]-W[o,c]; 0-pad.
// ---------------------------------------------------------------------------
__global__ void wprep_kernel(const float* __restrict__ W,
                             float* __restrict__ Wdp, float* __restrict__ Wcp,
                             int Cout, int Cin, int Cp) {
  int i = blockIdx.x * blockDim.x + threadIdx.x;
  if (i >= Cout * Cp) return;
  int o = i / Cp, c = i % Cp;
  float wd = 0.f, wc = 0.f;
  if (c < Cin) {
    float d = W[(long)o * 2 * Cin + c];
    wd = d;
    wc = W[(long)o * 2 * Cin + Cin + c] - d;
  }
  Wdp[i] = wd;
  Wcp[i] = wc;
}

// ---------------------------------------------------------------------------
// P[m, o] = sum_c A[m,c]*Bt[o,c] (+ bias[o]).  One wave per 16x16 tile.
// ---------------------------------------------------------------------------
template <int C>
__global__ __launch_bounds__(256)
void proj_kernel(const float* __restrict__ A, int lda,
                 const float* __restrict__ Bt, int ldb,
                 const float* __restrict__ bias,
                 float* __restrict__ P, int M, int Nout) {
  const int tilesN = Nout >> 4;
  const int wave = blockIdx.x * (blockDim.x >> 5) + (threadIdx.x >> 5);
  if (wave >= (M >> 4) * tilesN) return;
  const int tm = wave / tilesN, tn = wave % tilesN;

  v8f acc = wmma_nt_16x16<C>(A, lda, Bt, ldb, tm * 16, tn * 16);

  const int lane = threadIdx.x & 31;
  const int n = tn * 16 + (lane & 15);
  const int mbase = tm * 16 + 8 * (lane >> 4);
  const float bv = bias ? bias[n] : 0.f;
#pragma unroll
  for (int v = 0; v < 8; ++v)
    P[(long)(mbase + v) * Nout + n] = acc[v] + bv;
}

// ---------------------------------------------------------------------------
// EdgeConv epilogue: max_k lrelu(g*inv*(Dp[idx_k] + Ap) + be) -> F slice.
// ---------------------------------------------------------------------------
__global__ void edge_gather_max_kernel(const float* __restrict__ Dp,
                                       const float* __restrict__ Ap,
                                       const int* __restrict__ idx,
                                       const float* __restrict__ g,
                                       const float* __restrict__ be,
                                       float* __restrict__ Fout, int ldf,
                                       int N, int Cout, int Btot) {
  int i = blockIdx.x * blockDim.x + threadIdx.x;
  if (i >= Btot * N * Cout) return;
  const int o  = i % Cout;
  const int bn = i / Cout;
  const int b  = bn / N;
  const float scale = g[o] * INV_BN;
  const float shift = be[o];
  const float a = Ap[(long)bn * Cout + o];
  const int* ir = idx + (long)bn * KNN;
  float m = NEG_INF;
#pragma unroll
  for (int k = 0; k < KNN; ++k) {
    long j = (long)b * N + ir[k];
    float e = lrelu(scale * (Dp[j * Cout + o] + a) + shift);
    m = fmaxf(m, e);
  }
  Fout[(long)bn * ldf + o] = m;
}

// ---------------------------------------------------------------------------
// Fused final: out[b,o] = max_n lrelu(F[b,n,:512] . Wf[o,:] + bf[o]).
// Each wave owns 16 channels; per iteration processes 4 m-tiles sharing the
// weight fragment (1 W-load + 4 F-loads -> 4 WMMAs), max folds in registers.
// ---------------------------------------------------------------------------
__global__ __launch_bounds__(256)
void final_kernel(const float* __restrict__ F, const float* __restrict__ Wf,
                  const float* __restrict__ bf, float* __restrict__ out,
                  int N, int Cout) {
  const int b    = blockIdx.x >> 3;
  const int grp  = blockIdx.x & 7;
  const int wave = threadIdx.x >> 5;
  const int lane = threadIdx.x & 31;
  const int koff = (lane >> 4) * 2;
  const int l16  = lane & 15;
  const int n0   = grp * 128 + wave * 16;

  const float* Fb   = F  + (long)b * N * LDFEAT;
  const float* brow = Wf + (long)(n0 + l16) * LDFEAT + koff;
  const float  bias = bf[n0 + l16];
  float rm = NEG_INF;

  for (int tm = 0; tm < (N >> 4); tm += 4) {
    const float* a0 = Fb + (long)(tm * 16 + l16) * LDFEAT + koff;
    const float* a1 = a0 + 16 * LDFEAT;
    const float* a2 = a1 + 16 * LDFEAT;
    const float* a3 = a2 + 16 * LDFEAT;
    v8f c0 = {0.f,0.f,0.f,0.f,0.f,0.f,0.f,0.f};
    v8f c1 = c0, c2 = c0, c3 = c0;
#pragma unroll 2
    for (int k0 = 0; k0 < LDFEAT; k0 += 4) {
      v2f w  = *(const v2f*)(brow + k0);
      v2f x0 = *(const v2f*)(a0 + k0);
      v2f x1 = *(const v2f*)(a1 + k0);
      v2f x2 = *(const v2f*)(a2 + k0);
      v2f x3 = *(const v2f*)(a3 + k0);
      c0 = WMMA_F32(x0, w, c0);
      c1 = WMMA_F32(x1, w, c1);
      c2 = WMMA_F32(x2, w, c2);
      c3 = WMMA_F32(x3, w, c3);
    }
#pragma unroll
    for (int v = 0; v < 8; ++v) {
      float m01 = fmaxf(lrelu(c0[v] + bias), lrelu(c1[v] + bias));
      float m23 = fmaxf(lrelu(c2[v] + bias), lrelu(c3[v] + bias));
      rm = fmaxf(rm, fmaxf(m01, m23));
    }
  }
  rm = fmaxf(rm, __shfl_xor(rm, 16, 32));
  if (lane < 16) out[(long)b * Cout + n0 + lane] = rm;
}

// ---------------------------------------------------------------------------
// Host orchestration (stream-only, graph-capture safe, deterministic).
// Workspace (fp32 unless noted), ~204 MB:
//   F[8*4096*512] | dist[4096^2] | Dp[8*4096*256] | Ap[8*4096*256]
//   | sqn[8*4096] | Wdp[256*128] | Wcp[256*128] | Xp[8*4096*4]
//   | idxbuf[8*4096*20] (int)
// ---------------------------------------------------------------------------
extern "C" void kernel_launch(void* const* d_in, const int* in_sizes, int n_in,
                              void* d_out, int out_size, void* d_ws, size_t ws_size,
                              hipStream_t stream) {
  const float* x    = (const float*)d_in[0];
  const int*   idx0 = (const int*)  d_in[1];
  const float* Wf   = (const float*)d_in[18];
  const float* bf   = (const float*)d_in[19];
  float* out = (float*)d_out;

  float* F    = (float*)d_ws;
  float* dist = F    + (size_t)BATCH * NPTS * LDFEAT;
  float* Dp   = dist + (size_t)NPTS * NPTS;
  float* Ap   = Dp   + (size_t)BATCH * NPTS * 256;
  float* sqn  = Ap   + (size_t)BATCH * NPTS * 256;
  float* Wdp  = sqn  + (size_t)BATCH * NPTS;
  float* Wcp  = Wdp  + 256 * 128;
  float* Xp   = Wcp  + 256 * 128;
  int* idxbuf = (int*)(Xp + (size_t)BATCH * NPTS * 4);

  const int CinL[4]  = {3, 64, 64, 128};
  const int CpL[4]   = {4, 64, 64, 128};
  const int CoutL[4] = {64, 64, 128, 256};
  const int chOff[4] = {0, 64, 128, 256};
  const int M = BATCH * NPTS;

  padx_kernel<<<(M * 4 + 255) / 256, 256, 0, stream>>>(x, Xp, M);

  for (int li = 0; li < 4; ++li) {
    const int Ci = CinL[li], Cp = CpL[li], Co = CoutL[li];
    const float* W  = (const float*)d_in[2 + 4 * li];
    const float* bb = (const float*)d_in[3 + 4 * li];
    const float* g  = (const float*)d_in[4 + 4 * li];
    const float* be = (const float*)d_in[5 + 4 * li];

    const float* H; int ldh;
    if (li == 0) { H = Xp; ldh = 4; }
    else         { H = F + chOff[li - 1]; ldh = LDFEAT; }

    const int* idxp;
    if (li == 0) {
      idxp = idx0;
    } else {
      sqnorm_kernel<<<(M + 255) / 256, 256, 0, stream>>>(H, ldh, sqn, M, Ci);
      const int distBlocks = ((NPTS / 16) * (NPTS / 64)) / 8;    // 2048
      for (int b = 0; b < BATCH; ++b) {
        const float* Hb = H + (size_t)b * NPTS * ldh;
        const float* sb = sqn + b * NPTS;
        if (Ci == 64)
          knn_dist_kernel<64><<<distBlocks, 256, 0, stream>>>(Hb, ldh, sb, dist, NPTS);
        else
          knn_dist_kernel<128><<<distBlocks, 256, 0, stream>>>(Hb, ldh, sb, dist, NPTS);
        topk_kernel<<<NPTS / 8, 256, 0, stream>>>(dist, NPTS, idxbuf, b * NPTS);
      }
      idxp = idxbuf;
    }

    wprep_kernel<<<(Co * Cp + 255) / 256, 256, 0, stream>>>(W, Wdp, Wcp, Co, Ci, Cp);

    const int projBlocks = ((M / 16) * (Co / 16) + 7) / 8;
    if (Cp == 4) {
      proj_kernel<4><<<projBlocks, 256, 0, stream>>>(H, ldh, Wdp, Cp, nullptr, Dp, M, Co);
      proj_kernel<4><<<projBlocks, 256, 0, stream>>>(H, ldh, Wcp, Cp, bb,      Ap, M, Co);
    } else if (Cp == 64) {
      proj_kernel<64><<<projBlocks, 256, 0, stream>>>(H, ldh, Wdp, Cp, nullptr, Dp, M, Co);
      proj_kernel<64><<<projBlocks, 256, 0, stream>>>(H, ldh, Wcp, Cp, bb,      Ap, M, Co);
    } else {
      proj_kernel<128><<<projBlocks, 256, 0, stream>>>(H, ldh, Wdp, Cp, nullptr, Dp, M, Co);
      proj_kernel<128><<<projBlocks, 256, 0, stream>>>(H, ldh, Wcp, Cp, bb,      Ap, M, Co);
    }

    edge_gather_max_kernel<<<(M * Co + 255) / 256, 256, 0, stream>>>(
        Dp, Ap, idxp, g, be, F + chOff[li], LDFEAT, NPTS, Co, BATCH);
  }

  final_kernel<<<BATCH * 8, 256, 0, stream>>>(F, Wf, bf, out, NPTS, 1024);
}